// AdaIN_17712445129133
// MI455X (gfx1250) — compile-verified
//
#include <hip/hip_runtime.h>
#include <hip/hip_bf16.h>

typedef __attribute__((ext_vector_type(2))) float v2f;
typedef __attribute__((ext_vector_type(4))) float v4f;
typedef __attribute__((ext_vector_type(8))) float v8f;

#define NSEG 16
#define NCH  64

// workspace layout (floats)
#define C_SUM 0
#define C_SQ  1024
#define C_CNT 2048
#define S_SUM 2176
#define S_SQ  3200
#define S_CNT 4224
#define SCALE 4352
#define BIAS  5376
#define WS_FLOATS 6400

// ---------------------------------------------------------------------------
// Kernel 1: segment sum / sumsq / count via V_WMMA_F32_16X16X4_F32.
// sums[16x64] = onehot(idx)^T (16xN) * feats (Nx64), accumulated in VGPRs.
// Each wave handles 4 rows per WMMA group: A = 16x4 one-hot, B = 4x16 feats
// (one WMMA per 16-channel group, doubled for sum-of-squares).
// waveId is pinned to an SGPR (readfirstlane) so the loop is scalar-
// controlled: EXEC stays all-ones around the WMMAs and the latch is
// s_cmp/s_cbranch instead of exec-mask bookkeeping.
// ---------------------------------------------------------------------------
__global__ void __launch_bounds__(512)
seg_stats_wmma(const float* __restrict__ feats, const int* __restrict__ idx,
               int nrows,
               float* __restrict__ wsum, float* __restrict__ wsq,
               float* __restrict__ wcnt)
{
    __shared__ float lSum[NSEG * NCH];
    __shared__ float lSq[NSEG * NCH];
    __shared__ float lCnt[NSEG];

    for (int i = threadIdx.x; i < NSEG * NCH; i += blockDim.x) {
        lSum[i] = 0.0f;
        lSq[i]  = 0.0f;
    }
    if (threadIdx.x < NSEG) lCnt[threadIdx.x] = 0.0f;
    __syncthreads();

    const int lane = threadIdx.x & 31;
    const int m    = lane & 15;   // segment row of A / channel-in-group of B
    const int hi   = lane >> 4;   // half-wave selector

    // wave-uniform values, pinned to SGPRs
    const int waveId = __builtin_amdgcn_readfirstlane(
        (int)((blockIdx.x * blockDim.x + threadIdx.x) >> 5));
    const int nWaves = (gridDim.x * blockDim.x) >> 5;
    const int step   = nWaves * 4;

    v8f accS[4], accQ[4];
#pragma unroll
    for (int g = 0; g < 4; ++g) { accS[g] = (v8f){}; accQ[g] = (v8f){}; }
    float cnt = 0.0f;

    // -------- guard-free main loop over full 4-row groups --------
#pragma unroll 2
    for (int base = waveId * 4; base + 4 <= nrows; base += step) {
        // A fragment (16x4 one-hot):
        //   VGPR0 (A.x): K=0 (lanes 0-15) / K=2 (lanes 16-31)
        //   VGPR1 (A.y): K=1 (lanes 0-15) / K=3 (lanes 16-31)
        const int2 ss = *(const int2*)(idx + base + hi * 2);
        v2f A;
        A.x = (ss.x == m) ? 1.0f : 0.0f;
        A.y = (ss.y == m) ? 1.0f : 0.0f;
        cnt += A.x + A.y;   // each (row,seg) hit counted by exactly one lane

        // B fragments (4x16 per channel group):
        //   VGPR0: row base+hi, VGPR1: row base+2+hi; N = lane & 15
        const float* p0 = feats + (size_t)(base + hi) * NCH + m;
        const float* p1 = p0 + 2 * NCH;
        float b0[4], b1[4];
#pragma unroll
        for (int g = 0; g < 4; ++g) { b0[g] = p0[g * 16]; b1[g] = p1[g * 16]; }

#pragma unroll
        for (int g = 0; g < 4; ++g) {
            v2f B, B2;
            B.x  = b0[g];
            B.y  = b1[g];
            B2.x = b0[g] * b0[g];
            B2.y = b1[g] * b1[g];
            accS[g] = __builtin_amdgcn_wmma_f32_16x16x4_f32(
                false, A, false, B, (short)0, accS[g], false, false);
            accQ[g] = __builtin_amdgcn_wmma_f32_16x16x4_f32(
                false, A, false, B2, (short)0, accQ[g], false, false);
        }
    }

    // -------- flush accumulators into block-local LDS (ds_add_f32) --------
    // C/D layout: VGPR r -> seg = r + 8*hi, ch = g*16 + (lane&15)
    const int segLo = hi * 8;
#pragma unroll
    for (int g = 0; g < 4; ++g) {
        const int ch = g * 16 + m;
#pragma unroll
        for (int r = 0; r < 8; ++r) {
            const int o = (r + segLo) * NCH + ch;
            atomicAdd(&lSum[o], accS[g][r]);
            atomicAdd(&lSq[o],  accQ[g][r]);
        }
    }
    atomicAdd(&lCnt[m], cnt);

    // -------- scalar tail (< 4 rows), no WMMA, block 0 only --------
    if (blockIdx.x == 0 && threadIdx.x < NCH) {
        const int ch = threadIdx.x;
        for (int r = (nrows & ~3); r < nrows; ++r) {
            const int s   = idx[r];
            const float v = feats[(size_t)r * NCH + ch];
            atomicAdd(&lSum[s * NCH + ch], v);
            atomicAdd(&lSq[s * NCH + ch], v * v);
            if (ch == 0) atomicAdd(&lCnt[s], 1.0f);
        }
    }
    __syncthreads();

    // -------- one set of global atomics per block (low contention) --------
    for (int i = threadIdx.x; i < NSEG * NCH; i += blockDim.x) {
        atomicAdd(&wsum[i], lSum[i]);
        atomicAdd(&wsq[i],  lSq[i]);
    }
    if (threadIdx.x < NSEG) atomicAdd(&wcnt[threadIdx.x], lCnt[threadIdx.x]);
}

// ---------------------------------------------------------------------------
// Kernel 2: finalize stats + sequential 16-step EMA scan; emit fused
// scale/bias so the apply pass is a single FMA per element.
// One block of 64 threads (thread = channel).
// ---------------------------------------------------------------------------
__global__ void finalize_scan(const float* __restrict__ ws,
                              float* __restrict__ scale,
                              float* __restrict__ bias)
{
    const int ch = threadIdx.x;
    if (ch >= NCH) return;
    float gm = 0.0f, gs = 0.0f;
    for (int s = 0; s < NSEG; ++s) {
        const float ccnt  = ws[C_CNT + s];
        const float cmean = ws[C_SUM + s * NCH + ch] / ccnt;
        const float cvar  = (ws[C_SQ + s * NCH + ch] - ccnt * cmean * cmean)
                            / (ccnt - 1.0f);
        const float cstd  = sqrtf(fmaxf(cvar, 0.0f)) + 1e-8f;

        const float scnt  = ws[S_CNT + s];
        const float smean = ws[S_SUM + s * NCH + ch] / scnt;
        const float svar  = (ws[S_SQ + s * NCH + ch] - scnt * smean * smean)
                            / (scnt - 1.0f);
        const float sstd  = sqrtf(fmaxf(svar, 0.0f)) + 1e-8f;

        if (s == 0) { gm = smean; gs = sstd; }
        else        { gm = 0.9f * gm + 0.1f * smean;
                      gs = 0.9f * gs + 0.1f * sstd; }

        const float sc = gs / cstd;
        scale[s * NCH + ch] = sc;
        bias[s * NCH + ch]  = gm - cmean * sc;
    }
}

// ---------------------------------------------------------------------------
// Kernel 3: out = x * scale[seg][ch] + bias[seg][ch], 128-bit streaming.
// scale/bias cached in LDS (8 KB); non-temporal stores for the output
// stream; prefetch of the next grid-stride tile.
// ---------------------------------------------------------------------------
__global__ void __launch_bounds__(256)
adain_apply(const float* __restrict__ x, const int* __restrict__ idx,
            const float* __restrict__ scale, const float* __restrict__ bias,
            float* __restrict__ out, int nrows)
{
    __shared__ float sSc[NSEG * NCH];
    __shared__ float sBi[NSEG * NCH];
    for (int i = threadIdx.x; i < NSEG * NCH; i += blockDim.x) {
        sSc[i] = scale[i];
        sBi[i] = bias[i];
    }
    __syncthreads();

    const long total  = (long)nrows * (NCH / 4);   // one v4f per step
    const long stride = (long)gridDim.x * blockDim.x;
    for (long t = (long)blockIdx.x * blockDim.x + threadIdx.x; t < total;
         t += stride) {
        if (t + stride < total)
            __builtin_prefetch((const v4f*)x + t + stride, 0, 1);

        const long row = t >> 4;
        const int  cg  = (int)(t & 15);
        const int  seg = idx[row];
        const int  o   = seg * NCH + cg * 4;
        const v4f v = ((const v4f*)x)[t];
        v4f r;
        r.x = v.x * sSc[o + 0] + sBi[o + 0];
        r.y = v.y * sSc[o + 1] + sBi[o + 1];
        r.z = v.z * sSc[o + 2] + sBi[o + 2];
        r.w = v.w * sSc[o + 3] + sBi[o + 3];
        __builtin_nontemporal_store(r, (v4f*)out + t);
    }
}

// ---------------------------------------------------------------------------
extern "C" void kernel_launch(void* const* d_in, const int* in_sizes, int n_in,
                              void* d_out, int out_size, void* d_ws,
                              size_t ws_size, hipStream_t stream)
{
    const float* cf = (const float*)d_in[0];
    const float* sf = (const float*)d_in[1];
    const int*   ci = (const int*)d_in[2];
    const int*   si = (const int*)d_in[3];
    // num_batches is fixed at 16 == WMMA M dimension.

    const int nc = in_sizes[0] / NCH;
    const int ns = in_sizes[1] / NCH;

    float* ws = (float*)d_ws;
    (void)hipMemsetAsync(d_ws, 0, WS_FLOATS * sizeof(float), stream);

    seg_stats_wmma<<<256, 512, 0, stream>>>(cf, ci, nc,
                                            ws + C_SUM, ws + C_SQ, ws + C_CNT);
    seg_stats_wmma<<<96, 512, 0, stream>>>(sf, si, ns,
                                           ws + S_SUM, ws + S_SQ, ws + S_CNT);
    finalize_scan<<<1, 64, 0, stream>>>(ws, ws + SCALE, ws + BIAS);
    adain_apply<<<4096, 256, 0, stream>>>(cf, ci, ws + SCALE, ws + BIAS,
                                          (float*)d_out, nc);
}